// CodeFormerTorch_67224828117169
// MI455X (gfx1250) — compile-verified
//
#include <hip/hip_runtime.h>
#include <cstdint>

// AdaIN: B=16, C=512, H*W=4096, fp32.
// Memory-bound (≈400 MB @ 23.3 TB/s ≈ 17 us floor). One workgroup per (b,c)
// channel. CDNA5 path: async global->LDS copies (ASYNCcnt) stage the 16 KB
// slices, wave32 shuffle reductions for stats, register-resident second pass.

constexpr int   kSpatial  = 4096;        // H*W
constexpr int   kChannels = 16 * 512;    // B*C
constexpr int   kThreads  = 256;         // 8 wave32s
constexpr float kEps      = 1e-5f;

__device__ __forceinline__ uint32_t lds_addr_of(const void* p) {
  // Generic pointer into LDS aperture: low 32 bits == LDS byte offset
  return (uint32_t)(uintptr_t)p;
}

// GVS mode: mem_addr = SADDR(u64) + VADDR(i32). LDS dest addr per lane in VDST.
__device__ __forceinline__ void async_ld_b128(uint32_t lds_off, uint32_t voff,
                                              const float* sbase) {
  asm volatile("global_load_async_to_lds_b128 %0, %1, %2"
               :: "v"(lds_off), "v"(voff), "s"(sbase)
               : "memory");
}

__device__ __forceinline__ void wait_async_zero() {
  asm volatile("s_wait_asynccnt 0" ::: "memory");
}

__global__ __launch_bounds__(kThreads)
void adain_kernel(const float* __restrict__ soft,
                  const float* __restrict__ z,
                  float* __restrict__ out) {
  __shared__ __align__(16) float lsoft[kSpatial];   // 16 KB
  __shared__ __align__(16) float lz[kSpatial];      // 16 KB
  __shared__ float redS[8], redSS[8], redZ[8], redZZ[8];
  __shared__ float sScale, sShift;

  const int    tid  = threadIdx.x;
  const int    chan = blockIdx.x;
  const size_t base = (size_t)chan * kSpatial;
  const float* sChan = soft + base;
  const float* zChan = z + base;

  const uint32_t lsoft0 = lds_addr_of(lsoft);
  const uint32_t lz0    = lds_addr_of(lz);

  // Stage both 16 KB slices into LDS via the async copy engine.
  // Chunk k: lanes cover bytes [tid*16 + k*4096, +16) -- fully coalesced B128s.
#pragma unroll
  for (int k = 0; k < 4; ++k) {
    const uint32_t boff = (uint32_t)(tid * 16 + k * (kThreads * 16));
    async_ld_b128(lsoft0 + boff, boff, sChan);
    async_ld_b128(lz0 + boff, boff, zChan);
  }
  wait_async_zero();     // this wave's async loads have landed in LDS
  __syncthreads();       // all waves' loads visible

  // Pass 1: per-thread partial sum / sum^2 over 16 elements of each tensor.
  // Keep the soft values in registers for the normalize pass.
  float s_sum = 0.f, s_sq = 0.f, z_sum = 0.f, z_sq = 0.f;
  float4 sv[4];
#pragma unroll
  for (int k = 0; k < 4; ++k) {
    const int e = (tid + k * kThreads) * 4;
    const float4 s4 = *reinterpret_cast<const float4*>(&lsoft[e]);
    const float4 z4 = *reinterpret_cast<const float4*>(&lz[e]);
    sv[k] = s4;
    s_sum += (s4.x + s4.y) + (s4.z + s4.w);
    s_sq  += (s4.x * s4.x + s4.y * s4.y) + (s4.z * s4.z + s4.w * s4.w);
    z_sum += (z4.x + z4.y) + (z4.z + z4.w);
    z_sq  += (z4.x * z4.x + z4.y * z4.y) + (z4.z * z4.z + z4.w * z4.w);
  }

  // Wave32 butterfly reduction.
#pragma unroll
  for (int m = 16; m >= 1; m >>= 1) {
    s_sum += __shfl_xor(s_sum, m, 32);
    s_sq  += __shfl_xor(s_sq,  m, 32);
    z_sum += __shfl_xor(z_sum, m, 32);
    z_sq  += __shfl_xor(z_sq,  m, 32);
  }
  const int lane = tid & 31;
  const int wave = tid >> 5;
  if (lane == 0) {
    redS[wave]  = s_sum;
    redSS[wave] = s_sq;
    redZ[wave]  = z_sum;
    redZZ[wave] = z_sq;
  }
  __syncthreads();

  if (tid == 0) {
    float ts = 0.f, tss = 0.f, tz = 0.f, tzz = 0.f;
#pragma unroll
    for (int w = 0; w < 8; ++w) {
      ts += redS[w]; tss += redSS[w]; tz += redZ[w]; tzz += redZZ[w];
    }
    const float n    = (float)kSpatial;
    const float invn = 1.0f / n;
    const float mean_s = ts * invn;
    const float mean_z = tz * invn;
    // unbiased (ddof=1) variance, clamped against tiny negative from rounding
    const float var_s = fmaxf((tss - ts * ts * invn) / (n - 1.0f), 0.f);
    const float var_z = fmaxf((tzz - tz * tz * invn) / (n - 1.0f), 0.f);
    const float std_s = fmaxf(sqrtf(var_s), kEps);
    const float std_z = fmaxf(sqrtf(var_z), kEps);
    const float scale = std_z / std_s;
    sScale = scale;
    sShift = mean_z - mean_s * scale;
  }
  __syncthreads();

  const float scale = sScale;
  const float shift = sShift;

  // Pass 2: normalize from registers, coalesced float4 stores.
#pragma unroll
  for (int k = 0; k < 4; ++k) {
    const int e = (tid + k * kThreads) * 4;
    const float4 s4 = sv[k];
    float4 o;
    o.x = fmaf(s4.x, scale, shift);
    o.y = fmaf(s4.y, scale, shift);
    o.z = fmaf(s4.z, scale, shift);
    o.w = fmaf(s4.w, scale, shift);
    *reinterpret_cast<float4*>(&out[base + e]) = o;
  }
}

extern "C" void kernel_launch(void* const* d_in, const int* in_sizes, int n_in,
                              void* d_out, int out_size, void* d_ws, size_t ws_size,
                              hipStream_t stream) {
  (void)in_sizes; (void)n_in; (void)out_size; (void)d_ws; (void)ws_size;
  const float* soft = (const float*)d_in[0];
  const float* z    = (const float*)d_in[1];
  float*       out  = (float*)d_out;
  adain_kernel<<<kChannels, kThreads, 0, stream>>>(soft, z, out);
}